// DWPEDecomposition_80985903333616
// MI455X (gfx1250) — compile-verified
//
#include <hip/hip_runtime.h>
#include <cstdint>

// ---------------------------------------------------------------------------
// DWPE (8-level Haar wavelet packet + entropy gating) for MI455X / gfx1250.
//
// Identity: the 8-level Haar packet tree over each contiguous 256-sample block
// is a 256-point Walsh-Hadamard transform (x 1/16).  H256 = H16 (x) H16, so
// per block:  Y = H16 * X * H16  with X the block reshaped 16x16.  Each
// 16x16x16 f32 matmul = 4 chained V_WMMA_F32_16X16X4_F32 (exact in f32: H16
// entries are +/-1, generated in VALU via popcount).
// Leaf ordering: Hadamard row m -> output channel gray(bitrev8(m)).
//
// One workgroup (1024 thr = 32 wave32) per batch row; the full row of
// coefficients (256KB+pad) stays resident in CDNA5's 320KB LDS, so global
// traffic is the roofline minimum (~66MB total, ~3us at 23.3 TB/s).
// Operand chunks are preloaded into distinct registers so the 4-WMMA chains
// don't serialize on a single s_wait_loadcnt/dscnt per matrix op.
// ---------------------------------------------------------------------------

typedef __attribute__((ext_vector_type(2))) float v2f;
typedef __attribute__((ext_vector_type(8))) float v8f;

#define TLEN    65536
#define NCH     256
#define NK      256
#define NP      257          // padded rowbuf stride (bank decorrelation)
#define THREADS 1024
#define NWAVE   32
#define KPW     (NK / NWAVE) // 8 blocks per wave

__device__ __forceinline__ float hsign(int r, int c) {
    // H16[r][c] = (-1)^popcount(r & c)
    return (__popc(r & c) & 1) ? -1.0f : 1.0f;
}

extern "C" __global__ __launch_bounds__(THREADS)
void dwpe_row_kernel(const float* __restrict__ x,
                     float* __restrict__ outc,
                     float* __restrict__ oute,
                     float* __restrict__ outk) {
    extern __shared__ float smem[];
    float* rowbuf = smem;                      // NCH * NP   (full row of coeffs)
    float* twave  = rowbuf + NCH * NP;         // NWAVE * 256 (per-wave T tile)
    float* sums   = twave + NWAVE * 256;       // NCH
    float* kf     = sums + NCH;                // NCH

    const int b    = blockIdx.x;
    const int tid  = threadIdx.x;
    const int lane = tid & 31;
    const int wid  = tid >> 5;
    const int hi   = lane >> 4;   // 0: lanes 0-15, 1: lanes 16-31
    const int ln   = lane & 15;

    const float* xrow = x + (size_t)b * TLEN;
    float* tw = twave + wid * 256;

    // Loop-invariant +/-1 Hadamard operands (VALU-generated, hoisted).
    v2f hA[4], hB[4];
#pragma unroll
    for (int q = 0; q < 4; ++q) {
        const int k0 = 4 * q + 2 * hi;
        hA[q].x = hsign(ln, k0);      // A = H16 slice for GEMM 1
        hA[q].y = hsign(ln, k0 + 1);
        hB[q].x = hsign(k0, ln);      // B = H16 slice for GEMM 2
        hB[q].y = hsign(k0 + 1, ln);
    }

    for (int i = 0; i < KPW; ++i) {
        const int kblk = wid * KPW + i;
        const float* xb = xrow + kblk * 256;

        // ---- preload all B chunks (X rows) -> 8 loads in flight ----------
        v2f bm[4];
#pragma unroll
        for (int q = 0; q < 4; ++q) {
            const int k0 = 4 * q + 2 * hi;
            bm[q].x = xb[16 * k0 + ln];          // X[k0][ln]
            bm[q].y = xb[16 * (k0 + 1) + ln];    // X[k0+1][ln]
        }

        // ---- T = H16 * X (4 chained WMMA) --------------------------------
        v8f acc = {0.f,0.f,0.f,0.f,0.f,0.f,0.f,0.f};
#pragma unroll
        for (int q = 0; q < 4; ++q)
            acc = __builtin_amdgcn_wmma_f32_16x16x4_f32(
                false, hA[q], false, bm[q], (short)0, acc, false, false);

        // ---- stage T (C/D layout -> LDS) ---------------------------------
#pragma unroll
        for (int r = 0; r < 8; ++r)
            tw[(r + 8 * hi) * 16 + ln] = acc[r];
        asm volatile("" ::: "memory");  // keep stores before re-reads

        // ---- preload all A chunks of T (A layout) from LDS ---------------
        v2f am[4];
#pragma unroll
        for (int q = 0; q < 4; ++q) {
            const int k0 = 4 * q + 2 * hi;
            am[q].x = tw[ln * 16 + k0];          // T[ln][k0]
            am[q].y = tw[ln * 16 + k0 + 1];
        }

        // ---- Y = T * H16 (4 chained WMMA) --------------------------------
        v8f acc2 = {0.f,0.f,0.f,0.f,0.f,0.f,0.f,0.f};
#pragma unroll
        for (int q = 0; q < 4; ++q)
            acc2 = __builtin_amdgcn_wmma_f32_16x16x4_f32(
                false, am[q], false, hB[q], (short)0, acc2, false, false);

        // ---- scale, permute to frequency order, scatter into rowbuf ------
#pragma unroll
        for (int r = 0; r < 8; ++r) {
            const int m = 16 * (r + 8 * hi) + ln;           // Hadamard row
            const unsigned rev = __brev((unsigned)m) >> 24; // natural node id
            const int chan = (int)(rev ^ (rev >> 1));       // gray -> freq idx
            rowbuf[chan * NP + kblk] = acc2[r] * 0.0625f;   // 2^-4 Haar scale
        }
    }
    __syncthreads();

    // ---- per-channel sum of squares (4 lanes per channel) ----------------
    const int c    = tid >> 2;
    const int part = tid & 3;
    const float* rb = rowbuf + c * NP + part * 64;
    float s = 0.f;
    for (int j = 0; j < 64; ++j) { const float v = rb[j]; s += v * v; }
    s += __shfl_xor(s, 1);
    s += __shfl_xor(s, 2);
    if (part == 0) sums[c] = s;
    __syncthreads();

    // ---- per-channel entropy, keep mask ----------------------------------
    const float inv = 1.0f / (sums[c] + 1e-8f);
    float e = 0.f;
    for (int j = 0; j < 64; ++j) {
        const float v = rb[j];
        const float p = v * v * inv;
        e += p * __logf(p + 1e-8f);
    }
    e += __shfl_xor(e, 1);
    e += __shfl_xor(e, 2);
    if (part == 0) {
        const float ent  = -e;
        const float keep = (ent > 0.1f) ? 1.0f : 0.0f;
        kf[c] = keep;
        oute[b * NCH + c] = ent;
        outk[b * NCH + c] = keep;
    }
    __syncthreads();

    // ---- gated coefficient writeback (fully coalesced) -------------------
    float* oc = outc + (size_t)b * TLEN;
    for (int idx = tid; idx < TLEN; idx += THREADS) {
        const int cc = idx >> 8;
        const int k  = idx & 255;
        oc[idx] = rowbuf[cc * NP + k] * kf[cc];
    }
}

extern "C" void kernel_launch(void* const* d_in, const int* in_sizes, int n_in,
                              void* d_out, int out_size, void* d_ws, size_t ws_size,
                              hipStream_t stream) {
    (void)n_in; (void)out_size; (void)d_ws; (void)ws_size;
    const float* x = (const float*)d_in[0];
    const int B = in_sizes[0] / TLEN;

    float* outc = (float*)d_out;                       // [B,256,256]
    float* oute = outc + (size_t)B * TLEN;             // [B,256]
    float* outk = oute + (size_t)B * NCH;              // [B,256]

    const size_t lds = (size_t)(NCH * NP + NWAVE * 256 + 2 * NCH) * sizeof(float);
    // Opt in to >64KB dynamic LDS (CDNA5 WGP has 320KB). Deterministic, no
    // stream ops -> graph-capture safe.
    (void)hipFuncSetAttribute(reinterpret_cast<const void*>(dwpe_row_kernel),
                              hipFuncAttributeMaxDynamicSharedMemorySize,
                              (int)lds);
    dwpe_row_kernel<<<B, THREADS, lds, stream>>>(x, outc, oute, outk);
}